// _KSubspaceBaseModel_33603824124040
// MI455X (gfx1250) — compile-verified
//
#include <hip/hip_runtime.h>

#define KSUB 16
#define BATCH 2048
#define DIM 1024
#define DSUB 64

typedef __attribute__((ext_vector_type(2))) float v2f;
typedef __attribute__((ext_vector_type(8))) float v8f;

#define WMMA_F32(A, B, C) \
  __builtin_amdgcn_wmma_f32_16x16x4_f32(false, (A), false, (B), (short)0, (C), false, false)

// ---------------------------------------------------------------------------
// Kernel 0: zero the reduction scratch (losses, counts, losssum)
// ---------------------------------------------------------------------------
__global__ void zero_ws_kernel(float* __restrict__ p, int n) {
  int i = blockIdx.x * blockDim.x + threadIdx.x;
  if (i < n) p[i] = 0.0f;
}

// ---------------------------------------------------------------------------
// Kernel 1: z[k][b][j] = sum_D x[b][D] * Us[k][D][j]
// One wave32 computes a 16x64 strip (full d) with 4 f32 accumulators.
// A fragment (x rows) is loaded once per k-step and reused by 4 WMMAs.
// ---------------------------------------------------------------------------
__global__ __launch_bounds__(256) void gemm_z_kernel(
    const float* __restrict__ x, const float* __restrict__ Us,
    float* __restrict__ z) {
  const int lane = threadIdx.x & 31;
  const int wave = (blockIdx.x * blockDim.x + threadIdx.x) >> 5;

  const int k = wave / (BATCH / 16);
  const int btile = (wave % (BATCH / 16)) * 16;

  const int half = lane >> 4;  // 0: K 0/1, 1: K 2/3 (ISA 16x4 A/B layout)
  const int lid = lane & 15;   // A: row M, B: col N
  const int kbase = half * 2;

  const float* xrow = x + (size_t)(btile + lid) * DIM;
  const float* ub = Us + (size_t)k * DIM * DSUB + lid;  // + t*16 per col tile

  v8f c0 = {}, c1 = {}, c2 = {}, c3 = {};
#pragma unroll 2
  for (int kk = 0; kk < DIM; kk += 4) {
    v2f a = *(const v2f*)(xrow + kk + kbase);
    const float* u0 = ub + (size_t)(kk + kbase) * DSUB;
    const float* u1 = u0 + DSUB;
    v2f b0 = {u0[0], u1[0]};
    v2f b1 = {u0[16], u1[16]};
    v2f b2 = {u0[32], u1[32]};
    v2f b3 = {u0[48], u1[48]};
    c0 = WMMA_F32(a, b0, c0);
    c1 = WMMA_F32(a, b1, c1);
    c2 = WMMA_F32(a, b2, c2);
    c3 = WMMA_F32(a, b3, c3);
  }

  float* zt = z + (size_t)k * BATCH * DSUB + lid;
#pragma unroll
  for (int g = 0; g < 8; ++g) {
    const int m = g + 8 * half;  // C/D layout: VGPR g -> M=g (+8 hi half)
    float* row = zt + (size_t)(btile + m) * DSUB;
    row[0] = c0[g];
    row[16] = c1[g];
    row[32] = c2[g];
    row[48] = c3[g];
  }
}

// ---------------------------------------------------------------------------
// Kernel 2: x_[k][b][D] = sum_j z[k][b][j] * Us[k][D][j]   (z @ U^T)
// One wave32 computes a 16x64 slice of D: 16 kc-steps x 4 WMMAs, A reused.
// Fused squared-error: per-row partial over 64 columns, cross-lane reduce,
// one global f32 atomic per (row, k) per wave.
// ---------------------------------------------------------------------------
__global__ __launch_bounds__(256) void gemm_xr_kernel(
    const float* __restrict__ x, const float* __restrict__ Us,
    const float* __restrict__ z, float* __restrict__ xr,
    float* __restrict__ losses) {
  const int lane = threadIdx.x & 31;
  const int wave = (blockIdx.x * blockDim.x + threadIdx.x) >> 5;

  const int waves_per_k = (BATCH / 16) * (DIM / 64);  // 2048
  const int k = wave / waves_per_k;
  const int rem = wave % waves_per_k;
  const int btile = (rem / (DIM / 64)) * 16;
  const int ntile = (rem % (DIM / 64)) * 64;

  const int half = lane >> 4;
  const int lid = lane & 15;
  const int kbase = half * 2;

  const float* zrow = z + ((size_t)k * BATCH + btile + lid) * DSUB;
  // B[kc][n] = Us[k][n][kc] (U^T); per lane a contiguous v2f along kc
  const float* ur = Us + (size_t)k * DIM * DSUB;
  const float* u0 = ur + (size_t)(ntile + lid) * DSUB;
  const float* u1 = ur + (size_t)(ntile + 16 + lid) * DSUB;
  const float* u2 = ur + (size_t)(ntile + 32 + lid) * DSUB;
  const float* u3 = ur + (size_t)(ntile + 48 + lid) * DSUB;

  v8f c0 = {}, c1 = {}, c2 = {}, c3 = {};
#pragma unroll 4
  for (int kc = 0; kc < DSUB; kc += 4) {
    v2f a = *(const v2f*)(zrow + kc + kbase);
    v2f b0 = *(const v2f*)(u0 + kc + kbase);
    v2f b1 = *(const v2f*)(u1 + kc + kbase);
    v2f b2 = *(const v2f*)(u2 + kc + kbase);
    v2f b3 = *(const v2f*)(u3 + kc + kbase);
    c0 = WMMA_F32(a, b0, c0);
    c1 = WMMA_F32(a, b1, c1);
    c2 = WMMA_F32(a, b2, c2);
    c3 = WMMA_F32(a, b3, c3);
  }

  // store x_ strip and accumulate per-row squared error over 64 columns
  float sq[8];
#pragma unroll
  for (int g = 0; g < 8; ++g) {
    const int m = btile + g + 8 * half;
    float* orow = xr + (size_t)k * BATCH * DIM + (size_t)m * DIM + ntile + lid;
    const float* xrw = x + (size_t)m * DIM + ntile + lid;
    float acc = 0.0f;
    const float v0 = c0[g], v1 = c1[g], v2 = c2[g], v3 = c3[g];
    orow[0] = v0;
    orow[16] = v1;
    orow[32] = v2;
    orow[48] = v3;
    float d0 = xrw[0] - v0, d1 = xrw[16] - v1, d2 = xrw[32] - v2,
          d3 = xrw[48] - v3;
    acc = d0 * d0 + d1 * d1 + d2 * d2 + d3 * d3;
    sq[g] = acc;
  }

  // reduce across the 16 lanes of each half (xor masks < 16 stay in-half),
  // then one atomic per tile row.
#pragma unroll
  for (int g = 0; g < 8; ++g) {
    float v = sq[g];
    for (int mask = 1; mask < 16; mask <<= 1) v += __shfl_xor(v, mask, 32);
    if (lid == 0) {
      const int m = btile + g + 8 * half;
      atomicAdd(&losses[(size_t)m * KSUB + k], v);
    }
  }
}

// ---------------------------------------------------------------------------
// Kernel 3: per-sample argmin over K=16, one-hot c, usage/loss accumulators
// ---------------------------------------------------------------------------
__global__ void assign_kernel(const float* __restrict__ losses,
                              float* __restrict__ c_out,
                              float* __restrict__ counts,
                              float* __restrict__ losssum) {
  const int b = blockIdx.x * blockDim.x + threadIdx.x;
  if (b >= BATCH) return;
  float best = losses[b * KSUB];
  int bi = 0;
  for (int k = 1; k < KSUB; ++k) {
    const float v = losses[b * KSUB + k];
    if (v < best) {  // strict < : first occurrence, matches jnp.argmin
      best = v;
      bi = k;
    }
  }
  for (int k = 0; k < KSUB; ++k) c_out[b * KSUB + k] = (k == bi) ? 1.0f : 0.0f;
  atomicAdd(&counts[bi], 1.0f);
  atomicAdd(&losssum[bi], best);
}

// ---------------------------------------------------------------------------
// Kernel 4: EMA update + objective
// ---------------------------------------------------------------------------
__global__ void finalize_kernel(const float* __restrict__ c_mean,
                                const float* __restrict__ counts,
                                const float* __restrict__ losssum,
                                float* __restrict__ c_mean_new,
                                float* __restrict__ obj) {
  const int t = threadIdx.x;
  if (t < KSUB)
    c_mean_new[t] = 0.9f * c_mean[t] + 0.1f * (counts[t] / (float)BATCH);
  if (t == 0) {
    float s = 0.0f;
    for (int k = 0; k < KSUB; ++k) s += losssum[k];
    *obj = s / (float)BATCH;
  }
}

// ---------------------------------------------------------------------------
extern "C" void kernel_launch(void* const* d_in, const int* in_sizes, int n_in,
                              void* d_out, int out_size, void* d_ws,
                              size_t ws_size, hipStream_t stream) {
  const float* x = (const float*)d_in[0];       // (B, D)
  const float* Us = (const float*)d_in[1];      // (K, D, d)
  const float* c_mean = (const float*)d_in[2];  // (K,)

  float* out = (float*)d_out;
  float* out_xr = out;                              // (K, B, D)
  float* out_c = out + (size_t)KSUB * BATCH * DIM;  // (B, K)
  float* out_cm = out_c + (size_t)BATCH * KSUB;     // (K,)
  float* out_obj = out_cm + KSUB;                   // scalar

  float* ws = (float*)d_ws;
  float* zbuf = ws;                                    // (K, B, d)  8 MB
  float* losses = zbuf + (size_t)KSUB * BATCH * DSUB;  // (B, K)
  float* counts = losses + (size_t)BATCH * KSUB;       // (K,)
  float* losssum = counts + KSUB;                      // (K,)

  const int nzero = BATCH * KSUB + 2 * KSUB;
  zero_ws_kernel<<<(nzero + 255) / 256, 256, 0, stream>>>(losses, nzero);

  {
    const int waves = KSUB * (BATCH / 16);  // 2048 waves, 16x64 strips
    gemm_z_kernel<<<waves * 32 / 256, 256, 0, stream>>>(x, Us, zbuf);
  }
  {
    const int waves = KSUB * (BATCH / 16) * (DIM / 64);  // 32768 waves
    gemm_xr_kernel<<<waves * 32 / 256, 256, 0, stream>>>(x, Us, zbuf, out_xr,
                                                         losses);
  }
  assign_kernel<<<(BATCH + 255) / 256, 256, 0, stream>>>(losses, out_c, counts,
                                                         losssum);
  finalize_kernel<<<1, 32, 0, stream>>>(c_mean, counts, losssum, out_cm,
                                        out_obj);
}